// GCN_RNN_v2_87342454931923
// MI455X (gfx1250) — compile-verified
//
#include <hip/hip_runtime.h>
#include <hip/hip_bf16.h>

// ---------------------------------------------------------------------------
// GCN(2-layer) + LSTM pipeline for MI455X (gfx1250, wave32).
// All dense math uses V_WMMA_F32_16X16X4_F32 (fp32 in/out, matches reference
// precision). SpMM uses hardware global_atomic_add_f32.
//
// Round-3 fix: the hard 256-VGPR cap caused spills inside the WMMA loop.
// Instead, bound the scheduler's software-pipelining window structurally:
// partial-unroll (4) the LSTM k-loop so natural register pressure stays in
// the direct-file range. No hard allocator caps.
// ---------------------------------------------------------------------------

typedef float v2f __attribute__((ext_vector_type(2)));
typedef float v8f __attribute__((ext_vector_type(8)));

#define T_STEPS 8
#define NNODES  50000
#define NEDGES  800000
#define FDIM    128
#define HDIM    64
#define GATES   (4 * HDIM)   // 256
#define HPITCH  68           // h-tile LDS pitch (bank-conflict-free)

static __device__ __forceinline__ v8f wmma_f32(v2f a, v2f b, v8f c) {
  // (neg_a, A, neg_b, B, c_mod, C, reuse_a, reuse_b)
  return __builtin_amdgcn_wmma_f32_16x16x4_f32(false, a, false, b, (short)0, c,
                                               false, false);
}

static __device__ __forceinline__ float sigmoidf_(float x) {
  return 1.0f / (1.0f + __expf(-x));
}

static __device__ __forceinline__ float relu_(float x) {
  return (x > 0.f) ? x : 0.f;
}

// ---------------------------------------------------------------------------
// Zero a float buffer (float4 granularity).
// ---------------------------------------------------------------------------
__global__ __launch_bounds__(256) void zero_kernel(float4* __restrict__ p, long n4) {
  long i = (long)blockIdx.x * blockDim.x + threadIdx.x;
  if (i < n4) p[i] = make_float4(0.f, 0.f, 0.f, 0.f);
}

// ---------------------------------------------------------------------------
// GCN dense GEMM: out[M,64] = A[M,K] @ W[K,64], M = T*N = 400000.
// One wave computes a 16x64 tile via 16x16x4 f32 WMMA. W staged in LDS.
// Grid: 3125 blocks x 256 threads (8 waves) -> exactly 400000 rows.
// ---------------------------------------------------------------------------
template <int K>
__global__ __attribute__((amdgpu_waves_per_eu(4))) __launch_bounds__(256)
void gcn_gemm_kernel(const float* __restrict__ A, const float* __restrict__ W,
                     float* __restrict__ out) {
  __shared__ float wlds[K * HDIM];
  const int tid = threadIdx.x;
  for (int i = tid; i < K * HDIM; i += 256) wlds[i] = W[i];
  __syncthreads();

  const int lane  = tid & 31;
  const int wave  = tid >> 5;
  const int laneM = lane & 15;
  const int half  = lane >> 4;                 // 0: K-pair {0,1}, 1: K-pair {2,3}
  const long tileRow = (long)blockIdx.x * 8 + wave;

  const float* arow = A + (tileRow * 16 + laneM) * (long)K + half * 2;

  v8f acc[4] = {};
#pragma unroll
  for (int kb = 0; kb < K; kb += 4) {
    v2f a = *(const v2f*)(arow + kb);
    const int kr = kb + half * 2;
#pragma unroll
    for (int ct = 0; ct < 4; ++ct) {
      v2f b;
      b.x = wlds[kr * HDIM + ct * 16 + laneM];
      b.y = wlds[(kr + 1) * HDIM + ct * 16 + laneM];
      acc[ct] = wmma_f32(a, b, acc[ct]);
    }
  }

  float* orow = out + tileRow * 16 * HDIM;
#pragma unroll
  for (int ct = 0; ct < 4; ++ct)
#pragma unroll
    for (int r = 0; r < 8; ++r)
      orow[(r + half * 8) * HDIM + ct * 16 + laneM] = acc[ct][r];
}

// ---------------------------------------------------------------------------
// SpMM scatter-add: agg[t,row,:] += val * hw[t,col,:] for every edge, all t.
// One thread handles (edge, 2 h-values); edge triple loaded once, reused for
// all 8 time steps. fp32 HW atomics.
// Grid: E*32/256 = 100000 blocks.
// ---------------------------------------------------------------------------
__global__ __launch_bounds__(256) void spmm_kernel(
    const float* __restrict__ hw, const int* __restrict__ erow,
    const int* __restrict__ ecol, const float* __restrict__ eval,
    float* __restrict__ agg) {
  const long tid = (long)blockIdx.x * blockDim.x + threadIdx.x;
  const long e = tid >> 5;
  if (e >= NEDGES) return;
  const int hp = ((int)tid & 31) * 2;

  // Warm the L2 for the next edge batch (gfx1250 global_prefetch_b8).
  if (e + 2048 < NEDGES) {
    __builtin_prefetch(ecol + e + 2048, 0, 1);
    __builtin_prefetch(erow + e + 2048, 0, 1);
  }

  const int r = erow[e];
  const int c = ecol[e];
  const float v = eval[e];

  const long sl = (long)NNODES * HDIM;
  const float* src = hw + (long)c * HDIM + hp;
  float* dst = agg + (long)r * HDIM + hp;
#pragma unroll
  for (int t = 0; t < T_STEPS; ++t) {
    v2f x = *(const v2f*)(src + (long)t * sl);
    unsafeAtomicAdd(dst + (long)t * sl, v * x.x);
    unsafeAtomicAdd(dst + (long)t * sl + 1, v * x.y);
  }
}

// ---------------------------------------------------------------------------
// LSTM layer: one wave owns 16 nodes for the entire sequence.
//   gates = relu(xin[t]) @ Wih^T + h @ Whh^T + (bih+bhh)    (torch i,f,g,o)
// c kept in registers (WMMA D layout); h round-trips through a wave-private
// LDS tile (pitch 68 -> conflict-free) to become the next step's A operand.
// Both weight matrices staged in LDS, pair-transposed so every B fragment is
// a single aligned ds_load_b64:  w_p[kpair][j] = (W^T[2kp][j], W^T[2kp+1][j]).
// k-loop partially unrolled (4) to bound the scheduler's prefetch window.
// Output written with (stride_t, stride_n) so layer 1 emits [N,T,H] directly.
// Grid: 782 blocks x 128 threads (4 waves x 16 rows = 64 rows/block).
// LDS: 2*64KB weights + 4*4.25KB h-tiles = ~145KB (gfx1250: 320KB/WG max).
// ---------------------------------------------------------------------------
__global__ __attribute__((amdgpu_waves_per_eu(4))) __launch_bounds__(128)
void lstm_kernel(
    const float* __restrict__ xin,            // [T,N,64] pre-activation (relu here)
    const float* __restrict__ wih,            // [256,64]
    const float* __restrict__ whh,            // [256,64]
    const float* __restrict__ bih, const float* __restrict__ bhh,
    float* __restrict__ out, long stride_t, long stride_n) {
  __shared__ v2f wih_p[32 * GATES];           // [kpair][j]
  __shared__ v2f whh_p[32 * GATES];
  __shared__ float hlds_all[4][16 * HPITCH];

  const int tid  = threadIdx.x;
  // Stage weights pair-transposed: w_p[kp*256 + j] = (w[j*64+2kp], w[j*64+2kp+1])
  for (int idx = tid; idx < 32 * GATES; idx += 128) {
    const int kp = idx >> 8;                  // 0..31
    const int j  = idx & 255;                 // 0..255
    wih_p[idx] = *(const v2f*)(wih + (long)j * HDIM + kp * 2);
    whh_p[idx] = *(const v2f*)(whh + (long)j * HDIM + kp * 2);
  }

  const int lane  = tid & 31;
  const int wave  = tid >> 5;
  float* hlds = hlds_all[wave];
  const int laneM = lane & 15;
  const int half  = lane >> 4;

  const long tile  = (long)blockIdx.x * 4 + wave;
  const long m     = tile * 16 + laneM;
  const long mload = (m < NNODES) ? m : (NNODES - 1);   // clamp; WMMA needs full EXEC

  for (int i = lane; i < 16 * HPITCH; i += 32) hlds[i] = 0.f;

  // Per-lane bias gather, hoisted out of the time loop (16 scalars).
  float bias[16];
#pragma unroll
  for (int G = 0; G < 4; ++G)
#pragma unroll
    for (int ct = 0; ct < 4; ++ct) {
      const int j = G * HDIM + ct * 16 + laneM;
      bias[G * 4 + ct] = bih[j] + bhh[j];
    }

  __syncthreads();                            // weights visible to all waves

  v8f cst[4] = {};                            // cell state, D layout
  v2f ax[16];                                 // A fragments of relu(x_t)

  // Accumulate one gate group (16x64) = bias + x@Wih^T + h@Whh^T.
  auto compute_gates = [&](int G, v8f* acc) {
#pragma unroll
    for (int ct = 0; ct < 4; ++ct) {
      const float bj = bias[G * 4 + ct];
#pragma unroll
      for (int r = 0; r < 8; ++r) acc[ct][r] = bj;
    }
#pragma unroll 4
    for (int kk = 0; kk < 16; ++kk) {
      const int k  = kk * 4 + half * 2;
      const int kb = (kk * 2 + half) * GATES + G * HDIM;
      v2f ahf;                                // h fragment from LDS
      ahf.x = hlds[laneM * HPITCH + k];
      ahf.y = hlds[laneM * HPITCH + k + 1];
#pragma unroll
      for (int ct = 0; ct < 4; ++ct) {
        const int j = kb + ct * 16 + laneM;
        acc[ct] = wmma_f32(ax[kk], wih_p[j], acc[ct]);
        acc[ct] = wmma_f32(ahf,    whh_p[j], acc[ct]);
      }
    }
  };

  const long sl = (long)NNODES * HDIM;
  for (int t = 0; t < T_STEPS; ++t) {
    // --- load A fragments of relu(x_t) for this 16-row tile ---
    const float* xr = xin + (long)t * sl + mload * HDIM + half * 2;
#pragma unroll
    for (int kk = 0; kk < 16; ++kk) {
      v2f a = *(const v2f*)(xr + kk * 4);
      a.x = relu_(a.x);
      a.y = relu_(a.y);
      ax[kk] = a;
    }

    v8f acc[4];
    v8f si[4];
    // gate i
    compute_gates(0, acc);
#pragma unroll
    for (int ct = 0; ct < 4; ++ct)
#pragma unroll
      for (int r = 0; r < 8; ++r) si[ct][r] = sigmoidf_(acc[ct][r]);
    // gate f : c = sigmoid(f) * c
    compute_gates(1, acc);
#pragma unroll
    for (int ct = 0; ct < 4; ++ct)
#pragma unroll
      for (int r = 0; r < 8; ++r) cst[ct][r] = sigmoidf_(acc[ct][r]) * cst[ct][r];
    // gate g : c += sigmoid(i) * tanh(g)
    compute_gates(2, acc);
#pragma unroll
    for (int ct = 0; ct < 4; ++ct)
#pragma unroll
      for (int r = 0; r < 8; ++r) cst[ct][r] += si[ct][r] * tanhf(acc[ct][r]);
    // gate o : h = sigmoid(o) * tanh(c); publish h to LDS + global
    compute_gates(3, acc);
#pragma unroll
    for (int ct = 0; ct < 4; ++ct) {
#pragma unroll
      for (int r = 0; r < 8; ++r) {
        const float hv = sigmoidf_(acc[ct][r]) * tanhf(cst[ct][r]);
        const int rr = r + half * 8;
        hlds[rr * HPITCH + ct * 16 + laneM] = hv;
        const long grow = tile * 16 + rr;
        if (grow < NNODES)
          out[(long)t * stride_t + grow * stride_n + ct * 16 + laneM] = hv;
      }
    }
  }
}

// ---------------------------------------------------------------------------
// Host-side orchestration (graph-capture safe: only kernel launches).
// Workspace: hw [T,N,64] | agg [T,N,64] | hseq [T,N,64]  (~307 MB total)
// ---------------------------------------------------------------------------
extern "C" void kernel_launch(void* const* d_in, const int* in_sizes, int n_in,
                              void* d_out, int out_size, void* d_ws, size_t ws_size,
                              hipStream_t stream) {
  const float* x    = (const float*)d_in[0];
  const int*   erow = (const int*)d_in[1];
  const int*   ecol = (const int*)d_in[2];
  const float* eval = (const float*)d_in[3];
  const float* w0   = (const float*)d_in[4];
  const float* w1   = (const float*)d_in[5];
  const float* wih  = (const float*)d_in[6];   // [2,256,64]
  const float* whh  = (const float*)d_in[7];   // [2,256,64]
  const float* bih  = (const float*)d_in[8];   // [2,256]
  const float* bhh  = (const float*)d_in[9];   // [2,256]
  float* out = (float*)d_out;                  // [N,T,64]

  const size_t seq = (size_t)T_STEPS * NNODES * HDIM;
  float* hw   = (float*)d_ws;
  float* agg  = hw + seq;
  float* hseq = agg + seq;

  const long n4 = (long)seq / 4;
  const int zgrid = (int)((n4 + 255) / 256);
  const int gemm_blocks = (T_STEPS * NNODES) / 128;        // 3125
  const int spmm_blocks = (NEDGES * 32) / 256;             // 100000
  const int lstm_blocks = (NNODES + 63) / 64;              // 782

  // ---- layer 0 ----
  gcn_gemm_kernel<FDIM><<<gemm_blocks, 256, 0, stream>>>(x, w0, hw);
  zero_kernel<<<zgrid, 256, 0, stream>>>((float4*)agg, n4);
  spmm_kernel<<<spmm_blocks, 256, 0, stream>>>(hw, erow, ecol, eval, agg);
  lstm_kernel<<<lstm_blocks, 128, 0, stream>>>(
      agg, wih, whh, bih, bhh, hseq,
      /*stride_t=*/(long)NNODES * HDIM, /*stride_n=*/(long)HDIM);

  // ---- layer 1 ----
  gcn_gemm_kernel<HDIM><<<gemm_blocks, 256, 0, stream>>>(hseq, w1, hw);
  zero_kernel<<<zgrid, 256, 0, stream>>>((float4*)agg, n4);
  spmm_kernel<<<spmm_blocks, 256, 0, stream>>>(hw, erow, ecol, eval, agg);
  lstm_kernel<<<lstm_blocks, 128, 0, stream>>>(
      agg, wih + (size_t)GATES * HDIM, whh + (size_t)GATES * HDIM,
      bih + GATES, bhh + GATES, out,
      /*stride_t=*/(long)HDIM, /*stride_n=*/(long)T_STEPS * HDIM);
}